// MVDR_25142738550941
// MI455X (gfx1250) — compile-verified
//
#include <hip/hip_runtime.h>

// Problem constants (from reference setup_inputs)
#define BB 4
#define CC 6
#define FF 257
#define TT 2000
#define BF (BB * FF)      // 1028 (b,f) bins
#define SEG 10            // split T into 10 segments of 200 for parallelism
#define TSEG (TT / SEG)   // 200, multiple of 8 (2x unrolled WMMA K=4)

typedef __attribute__((ext_vector_type(2))) float v2f;
typedef __attribute__((ext_vector_type(8))) float v8f;

// ---------------------------------------------------------------------------
// Kernel 1: per-(b,f) mask sums -> inverse normalizers 1/(sum + 1e-15)
// ---------------------------------------------------------------------------
__global__ void mvdr_mask_norm(const float* __restrict__ mask_s,
                               const float* __restrict__ mask_n,
                               float* __restrict__ inv_s,
                               float* __restrict__ inv_n) {
  int bf = blockIdx.x;
  const float* ms = mask_s + (size_t)bf * TT;
  const float* mn = mask_n + (size_t)bf * TT;
  float ss = 0.f, sn = 0.f;
  for (int t = threadIdx.x; t < TT; t += blockDim.x) { ss += ms[t]; sn += mn[t]; }
  __shared__ float rs[256], rn[256];
  rs[threadIdx.x] = ss; rn[threadIdx.x] = sn;
  __syncthreads();
  for (int off = 128; off > 0; off >>= 1) {
    if ((int)threadIdx.x < off) {
      rs[threadIdx.x] += rs[threadIdx.x + off];
      rn[threadIdx.x] += rn[threadIdx.x + off];
    }
    __syncthreads();
  }
  if (threadIdx.x == 0) {
    inv_s[bf] = 1.0f / (rs[0] + 1e-15f);
    inv_n[bf] = 1.0f / (rn[0] + 1e-15f);
  }
}

// ---------------------------------------------------------------------------
// Kernel 2: PSD Gram matrices via V_WMMA_F32_16X16X4_F32.
// One wave per (bf, seg). A rows 0..5 = Re channels, 6..11 = Im channels,
// rows 12..15 = clamped duplicates (they only pollute G rows/cols >= 12,
// which are never read -> no zeroing, no divergence).
// B = m(t) * A^T. Per 8 t's: one b128 load per stream, 4 WMMAs.
// K-slot permutation trick: the Gram sum is order-independent over t, so
// WMMA #1 takes K slots {kb,kb+1} from half 0 and {kb+4,kb+5} from half 1
// (the .x/.y of each half's float4), WMMA #2 takes the .z/.w pairs.
// C/D layout: VGPR j holds row (j + 8*half), col = lane&15.
// ---------------------------------------------------------------------------
__global__ void mvdr_psd_wmma(const float* __restrict__ spec_r,
                              const float* __restrict__ spec_i,
                              const float* __restrict__ mask_s,
                              const float* __restrict__ mask_n,
                              const float* __restrict__ inv_s,
                              const float* __restrict__ inv_n,
                              float* __restrict__ Gout) {
  int wave = threadIdx.x >> 5;          // 8 waves / block
  int lane = threadIdx.x & 31;
  int work = blockIdx.x * 8 + wave;     // == bf*SEG + seg
  if (work >= BF * SEG) return;         // wave-uniform guard
  int bf  = work / SEG;
  int seg = work - bf * SEG;
  int b = bf / FF;
  int f = bf - b * FF;
  int half = lane >> 4;
  int m    = lane & 15;

  // rows 0..5 -> Re ch m, rows 6..11 -> Im ch m-6, rows 12..15 -> Re ch m-12
  int ch = (m < 6) ? m : ((m < 12) ? (m - 6) : (m - 12));
  const float* base   = (m >= 6 && m < 12) ? spec_i : spec_r;
  const float* rowptr = base + (size_t)(((b * CC + ch) * FF) + f) * TT;
  const float* msrow  = mask_s + (size_t)bf * TT;
  const float* mnrow  = mask_n + (size_t)bf * TT;
  float is  = inv_s[bf];
  float in_ = inv_n[bf];

  v8f accS = {0.f,0.f,0.f,0.f,0.f,0.f,0.f,0.f};
  v8f accN = {0.f,0.f,0.f,0.f,0.f,0.f,0.f,0.f};

  int t0 = seg * TSEG;
  int ho = 4 * half;                    // half-1 lanes take t+4..t+7
  for (int kb = t0; kb < t0 + TSEG; kb += 8) {
    float4 a4 = *(const float4*)(rowptr + kb + ho);   // 16B aligned
    float4 s4 = *(const float4*)(msrow  + kb + ho);
    float4 n4 = *(const float4*)(mnrow  + kb + ho);

    v2f a01  = {a4.x, a4.y};
    v2f a23  = {a4.z, a4.w};
    v2f bs01 = {a4.x * s4.x * is,  a4.y * s4.y * is };
    v2f bs23 = {a4.z * s4.z * is,  a4.w * s4.w * is };
    v2f bn01 = {a4.x * n4.x * in_, a4.y * n4.y * in_};
    v2f bn23 = {a4.z * n4.z * in_, a4.w * n4.w * in_};

    // D = A x B + C  (f32 16x16x4); interleave S/N chains for ILP
    accS = __builtin_amdgcn_wmma_f32_16x16x4_f32(false, a01, false, bs01, (short)0, accS, false, false);
    accN = __builtin_amdgcn_wmma_f32_16x16x4_f32(false, a01, false, bn01, (short)0, accN, false, false);
    accS = __builtin_amdgcn_wmma_f32_16x16x4_f32(false, a23, false, bs23, (short)0, accS, false, false);
    accN = __builtin_amdgcn_wmma_f32_16x16x4_f32(false, a23, false, bn23, (short)0, accN, false, false);
  }

  float* G = Gout + (size_t)work * 512;   // 256 floats (S) + 256 floats (N)
  int row0 = half * 8;
#pragma unroll
  for (int j = 0; j < 8; ++j) {
    G[(row0 + j) * 16 + m]       = accS[j];
    G[256 + (row0 + j) * 16 + m] = accN[j];
  }
}

// ---------------------------------------------------------------------------
// Kernel 3: per-(b,f): sum segment partials, assemble complex 6x6 PSDs,
// Tikhonov-load psd_n, invert (Gauss-Jordan), trace-normalize, emit conj(w).
// ---------------------------------------------------------------------------
__device__ __forceinline__ void cmul(float ar, float ai, float br, float bi,
                                     float& cr, float& ci) {
  cr = ar * br - ai * bi;
  ci = ar * bi + ai * br;
}

__global__ void mvdr_solve(const float* __restrict__ Gin,
                           float* __restrict__ wout) {
  int idx = blockIdx.x * blockDim.x + threadIdx.x;
  if (idx >= BF) return;

  float Sr[36], Si[36], Nr[36], Ni[36];
#pragma unroll
  for (int c = 0; c < 6; ++c) {
#pragma unroll
    for (int e = 0; e < 6; ++e) {
      float s00 = 0.f, s11 = 0.f, s10 = 0.f, s01 = 0.f;
      float n00 = 0.f, n11 = 0.f, n10 = 0.f, n01 = 0.f;
      for (int g = 0; g < SEG; ++g) {
        const float* G = Gin + (size_t)(idx * SEG + g) * 512;
        s00 += G[c * 16 + e];
        s11 += G[(c + 6) * 16 + (e + 6)];
        s10 += G[(c + 6) * 16 + e];
        s01 += G[c * 16 + (e + 6)];
        n00 += G[256 + c * 16 + e];
        n11 += G[256 + (c + 6) * 16 + (e + 6)];
        n10 += G[256 + (c + 6) * 16 + e];
        n01 += G[256 + c * 16 + (e + 6)];
      }
      Sr[c * 6 + e] = s00 + s11;  Si[c * 6 + e] = s10 - s01;
      Nr[c * 6 + e] = n00 + n11;  Ni[c * 6 + e] = n10 - n01;
    }
  }

  // Tikhonov diagonal loading: eps = tr(psd_n)*1e-7 + 1e-8
  float trn = 0.f;
#pragma unroll
  for (int c = 0; c < 6; ++c) trn += Nr[c * 7];
  float eps = trn * 1e-7f + 1e-8f;
#pragma unroll
  for (int c = 0; c < 6; ++c) Nr[c * 7] += eps;

  // Gauss-Jordan inverse of N (Hermitian PD after loading -> diagonal pivots OK)
  float Jr[36], Ji[36];
#pragma unroll
  for (int i = 0; i < 36; ++i) { Jr[i] = 0.f; Ji[i] = 0.f; }
#pragma unroll
  for (int i = 0; i < 6; ++i) Jr[i * 7] = 1.f;

#pragma unroll
  for (int k = 0; k < 6; ++k) {
    float pr = Nr[k * 6 + k], pi = Ni[k * 6 + k];
    float d  = pr * pr + pi * pi + 1e-30f;
    float ir = pr / d, ii = -pi / d;        // 1/pivot
#pragma unroll
    for (int j = 0; j < 6; ++j) {
      float tr0, ti0;
      cmul(Nr[k * 6 + j], Ni[k * 6 + j], ir, ii, tr0, ti0);
      Nr[k * 6 + j] = tr0; Ni[k * 6 + j] = ti0;
      cmul(Jr[k * 6 + j], Ji[k * 6 + j], ir, ii, tr0, ti0);
      Jr[k * 6 + j] = tr0; Ji[k * 6 + j] = ti0;
    }
#pragma unroll
    for (int i = 0; i < 6; ++i) {
      if (i == k) continue;
      float fr = Nr[i * 6 + k], fi = Ni[i * 6 + k];
#pragma unroll
      for (int j = 0; j < 6; ++j) {
        float tr0, ti0;
        cmul(fr, fi, Nr[k * 6 + j], Ni[k * 6 + j], tr0, ti0);
        Nr[i * 6 + j] -= tr0; Ni[i * 6 + j] -= ti0;
        cmul(fr, fi, Jr[k * 6 + j], Ji[k * 6 + j], tr0, ti0);
        Jr[i * 6 + j] -= tr0; Ji[i * 6 + j] -= ti0;
      }
    }
  }

  // tr(inv(N)*S) and w_un = inv(N)*S[:,0]
  float trr = 0.f, tri = 0.f;
  float wur[6], wui[6];
#pragma unroll
  for (int c = 0; c < 6; ++c) {
    float ar = 0.f, ai = 0.f;
#pragma unroll
    for (int e = 0; e < 6; ++e) {
      float tr0, ti0;
      cmul(Jr[c * 6 + e], Ji[c * 6 + e], Sr[e * 6 + c], Si[e * 6 + c], tr0, ti0);
      trr += tr0; tri += ti0;
      cmul(Jr[c * 6 + e], Ji[c * 6 + e], Sr[e * 6 + 0], Si[e * 6 + 0], tr0, ti0);
      if (e == 0) { ar = tr0; ai = ti0; } else { ar += tr0; ai += ti0; }
    }
    wur[c] = ar; wui[c] = ai;
  }

  // w = w_un / (tr + 1e-8); store conj(w)
  float dr = trr + 1e-8f, di = tri;
  float dd = dr * dr + di * di + 1e-30f;
#pragma unroll
  for (int c = 0; c < 6; ++c) {
    float wr = (wur[c] * dr + wui[c] * di) / dd;
    float wi = (wui[c] * dr - wur[c] * di) / dd;
    wout[idx * 12 + 2 * c + 0] = wr;    // Re(conj(w)) =  Re(w)
    wout[idx * 12 + 2 * c + 1] = -wi;   // Im(conj(w)) = -Im(w)
  }
}

// ---------------------------------------------------------------------------
// Kernel 4: streaming beamform  enh[t] = sum_c conj(w)[c] * spec[c,t]
// out layout: (B,F,T,2) -> float2 per t
// ---------------------------------------------------------------------------
__global__ void mvdr_beamform(const float* __restrict__ spec_r,
                              const float* __restrict__ spec_i,
                              const float* __restrict__ wvec,
                              float* __restrict__ out) {
  int bf = blockIdx.x;
  int t  = blockIdx.y * blockDim.x + threadIdx.x;
  if (t >= TT) return;
  int b = bf / FF;
  int f = bf - b * FF;
  float outr = 0.f, outi = 0.f;
#pragma unroll
  for (int c = 0; c < CC; ++c) {
    float ar = wvec[bf * 12 + 2 * c + 0];   // block-uniform -> scalar loads
    float ai = wvec[bf * 12 + 2 * c + 1];
    size_t off = (size_t)(((b * CC + c) * FF) + f) * TT + t;
    float R = spec_r[off];
    float I = spec_i[off];
    outr += ar * R - ai * I;
    outi += ar * I + ai * R;
  }
  float2* o = (float2*)out + (size_t)bf * TT + t;
  *o = make_float2(outr, outi);
}

// ---------------------------------------------------------------------------
extern "C" void kernel_launch(void* const* d_in, const int* in_sizes, int n_in,
                              void* d_out, int out_size, void* d_ws, size_t ws_size,
                              hipStream_t stream) {
  (void)in_sizes; (void)n_in; (void)out_size; (void)ws_size;
  const float* spec_r = (const float*)d_in[0];
  const float* spec_i = (const float*)d_in[1];
  const float* mask_s = (const float*)d_in[2];
  const float* mask_n = (const float*)d_in[3];
  float* out = (float*)d_out;

  // workspace layout (floats): inv_s[BF] | inv_n[BF] | G[BF*SEG*512] | w[BF*12]
  float* ws    = (float*)d_ws;
  float* inv_s = ws;
  float* inv_n = ws + BF;
  float* G     = ws + 2 * BF;
  float* wvec  = G + (size_t)BF * SEG * 512;

  mvdr_mask_norm<<<BF, 256, 0, stream>>>(mask_s, mask_n, inv_s, inv_n);

  int nwork = BF * SEG;                         // 10280 waves
  mvdr_psd_wmma<<<(nwork + 7) / 8, 256, 0, stream>>>(spec_r, spec_i, mask_s, mask_n,
                                                     inv_s, inv_n, G);

  mvdr_solve<<<(BF + 63) / 64, 64, 0, stream>>>(G, wvec);

  mvdr_beamform<<<dim3(BF, (TT + 255) / 256), 256, 0, stream>>>(spec_r, spec_i, wvec, out);
}